// CapsuleLayer_11441792877351
// MI455X (gfx1250) — compile-verified
//
#include <hip/hip_runtime.h>
#include <hip/hip_bf16.h>

// CDNA5 / gfx1250 capsule-routing kernel.
// Key idea: b_logits are linear in u_hat (b0 = 0, additive updates), so
//   b_r[b,n,i] = dot(V_r[b,n,:], u_hat[b,n,i,:]),  V_r = sum_{r'<r} v_{r'}.
// We therefore never materialize the 189MB u_hat tensor: each routing pass
// recomputes u_hat tiles with V_WMMA_F32_16X16X4_F32 straight out of the
// L2-resident inputs (9.4MB) and W (5.9MB).

typedef __attribute__((ext_vector_type(2))) float v2f;
typedef __attribute__((ext_vector_type(4))) float v4f;
typedef __attribute__((ext_vector_type(8))) float v8f;

#define CB   256          // batch
#define CN   10           // output capsules
#define CI   1152         // input capsules
#define CD   16           // output capsule dim
#define CDIN 8            // input capsule dim
#define BT   16           // batch tile (WMMA N dim)
#define NSLICE 18         // i-dimension slices per (b-tile)
#define ITILE (CI / NSLICE) // 64

static_assert(CI % NSLICE == 0, "i slices must divide I");

// ---------------------------------------------------------------------------
// One routing pass. Grid: (CB/BT, NSLICE). Block: 10 waves (320 threads),
// wave w handles capsule n = w so the per-i softmax over n is a cheap LDS
// exchange of 10 scalars per batch column.
//
// WMMA tile (f32 16x16x4, chained twice for K=8):
//   C[d(16) x b(16)] = A(W[n,i,d,k]) x B(inputs[b,i,k])
// A layout (f32 16x4): lane L row M=L%16; lanes<16 hold K=0,1; lanes>=16 K=2,3
// B layout (f32 4x16): mirrored K striping across lane halves
// C layout: acc[r] on lane L = u_hat[d = r + 8*(L>=16)][b = L%16]
// ---------------------------------------------------------------------------
__global__ __launch_bounds__(32 * CN) void caps_pass_kernel(
    const float* __restrict__ X,   // [B][I][CDIN]
    const float* __restrict__ W,   // [N][I][CD][CDIN]
    const float* __restrict__ V,   // [B][N][CD] running sum of past outputs
    float* __restrict__ S)         // [B][N][CD] s accumulator (pre-zeroed)
{
    __shared__ float logits[2][CN][BT];   // double-buffered -> 1 barrier/iter

    const int lane = threadIdx.x & 31;
    const int n    = threadIdx.x >> 5;    // capsule owned by this wave
    const int l16  = lane & 15;           // A-row d index / B-col b index
    const int kh   = lane >> 4;           // K-half selector (WMMA striping)
    const int b0   = blockIdx.x * BT;
    const int i0   = blockIdx.y * ITILE;

    // Per-lane slice of V: V[b0+l16][n][kh*8 + r], r = 0..7 (matches C rows)
    const float* vp = V + (((b0 + l16) * CN + n) * CD) + kh * 8;
    const v4f vlo = *(const v4f*)(vp);
    const v4f vhi = *(const v4f*)(vp + 4);

    float sacc[8] = {0.f, 0.f, 0.f, 0.f, 0.f, 0.f, 0.f, 0.f};

    for (int ii = 0; ii < ITILE; ++ii) {
        const int i = i0 + ii;

        // A operand: W[n][i][d = l16][k]; rows are 8 contiguous f32 (32B aligned)
        const float* wr = W + (((n * CI + i) * CD) + l16) * CDIN;
        const v2f a0 = *(const v2f*)(wr + 2 * kh);      // k = 2h, 2h+1
        const v2f a1 = *(const v2f*)(wr + 4 + 2 * kh);  // k = 4+2h, 5+2h

        // B operand: inputs[b = l16][i][k]
        const float* xr = X + (((b0 + l16) * CI + i) * CDIN);
        const v2f q0 = *(const v2f*)(xr + 2 * kh);
        const v2f q1 = *(const v2f*)(xr + 4 + 2 * kh);

        // u_hat tile: two chained K=4 f32 WMMAs (exact f32 semantics)
        v8f acc = {};
        acc = __builtin_amdgcn_wmma_f32_16x16x4_f32(
                  false, a0, false, q0, (short)0, acc, false, false);
        acc = __builtin_amdgcn_wmma_f32_16x16x4_f32(
                  false, a1, false, q1, (short)0, acc, false, false);

        // routing logit = dot(V[b,n,:], u_hat[b,n,i,:]):
        // 8 in-register FMAs + one cross-half shuffle (d split across halves)
        float t = vlo[0]*acc[0] + vlo[1]*acc[1] + vlo[2]*acc[2] + vlo[3]*acc[3]
                + vhi[0]*acc[4] + vhi[1]*acc[5] + vhi[2]*acc[6] + vhi[3]*acc[7];
        t += __shfl_xor(t, 16);

        const int par = ii & 1;
        if (lane < 16) logits[par][n][l16] = t;
        __syncthreads();

        // softmax over the 10 capsules for this (b, i); we need c[n] only
        float l[CN];
        float m = -1e30f;
        #pragma unroll
        for (int q = 0; q < CN; ++q) {
            l[q] = logits[par][q][l16];
            m = fmaxf(m, l[q]);
        }
        float den = 0.f, en = 0.f;
        #pragma unroll
        for (int q = 0; q < CN; ++q) {
            const float e = __expf(l[q] - m);
            den += e;
            if (q == n) en = e;
        }
        const float c = en / den;

        // s[b,n,d] += c * u_hat[b,n,i,d] (register-resident)
        #pragma unroll
        for (int r = 0; r < 8; ++r) sacc[r] += c * acc[r];
    }

    // Flush partial s; slices along grid-y collide -> f32 global atomics.
    float* sp = S + (((b0 + l16) * CN + n) * CD) + kh * 8;
    #pragma unroll
    for (int r = 0; r < 8; ++r) atomicAdd(sp + r, sacc[r]);
}

// ---------------------------------------------------------------------------
// squash(s) = s * ||s|| / (1 + ||s||^2); accumulate into V, or emit output.
// 16 lanes per (b,n) group; norm via 4 xor-shuffles inside the group.
// ---------------------------------------------------------------------------
__global__ __launch_bounds__(256) void caps_squash_kernel(
    const float* __restrict__ S,
    float* __restrict__ V,
    float* __restrict__ out,
    int is_last)
{
    const int g = blockIdx.x * blockDim.x + threadIdx.x;  // exact grid: B*N*D
    const float s = S[g];
    float p = s * s;
    p += __shfl_xor(p, 1);
    p += __shfl_xor(p, 2);
    p += __shfl_xor(p, 4);
    p += __shfl_xor(p, 8);
    const float ns  = p;
    const float v   = s * (sqrtf(ns) / (1.0f + ns));
    if (is_last) out[g] = v;
    else         V[g] += v;
}

extern "C" void kernel_launch(void* const* d_in, const int* in_sizes, int n_in,
                              void* d_out, int out_size, void* d_ws, size_t ws_size,
                              hipStream_t stream) {
    (void)in_sizes; (void)n_in; (void)out_size; (void)ws_size;
    const float* X = (const float*)d_in[0];   // [256,1152,8]  f32
    const float* W = (const float*)d_in[1];   // [10,1152,16,8] f32
    float* out = (float*)d_out;               // [256,10,16]   f32

    float* V = (float*)d_ws;                  // 40960 f32 = 160 KB
    float* S = V + CB * CN * CD;              // 40960 f32 = 160 KB

    hipMemsetAsync(V, 0, CB * CN * CD * sizeof(float), stream);

    const dim3 gridA(CB / BT, NSLICE);        // 16 x 18 = 288 workgroups
    for (int r = 0; r < 3; ++r) {
        hipMemsetAsync(S, 0, CB * CN * CD * sizeof(float), stream);
        caps_pass_kernel<<<gridA, 32 * CN, 0, stream>>>(X, W, V, S);
        caps_squash_kernel<<<(CB * CN * CD) / 256, 256, 0, stream>>>(
            S, V, out, r == 2 ? 1 : 0);
    }
}